// GIN_74904229642495
// MI455X (gfx1250) — compile-verified
//
#include <hip/hip_runtime.h>

// ---------------------------------------------------------------------------
// GIN forward (3 layers) for MI455X / gfx1250.
//   agg = segment_sum(x[src] -> dst);  h = (1+eps)*x + agg;  out = h@W + b
// Layers 1,2 apply ReLU; layer 3 does not.
// ---------------------------------------------------------------------------

#define N_NODES 50000
#define N_EDGES 800000
#define IN_CH   128
#define HID_CH  256
#define OUT_CH  256

static_assert(N_NODES % 16 == 0, "M tiles must be exact");

typedef __attribute__((ext_vector_type(2))) float v2f;
typedef __attribute__((ext_vector_type(8))) float v8f;

// ---------------------------------------------------------------------------
// Zero-fill (d_ws is poisoned by the harness; agg must be zeroed every call)
// ---------------------------------------------------------------------------
__global__ __launch_bounds__(256)
void fill_zero(float* __restrict__ p, long long n) {
  long long i = (long long)blockIdx.x * blockDim.x + threadIdx.x;
  const long long stride = (long long)gridDim.x * blockDim.x;
  for (; i < n; i += stride) p[i] = 0.0f;
}

// ---------------------------------------------------------------------------
// Scatter-add:  agg[dst[e]*C + c] += x[src[e]*C + c]
// One thread per (edge, 4-channel group): float4 coalesced read, 4 HW f32
// atomics (global_atomic_add_f32, no-return -> STOREcnt path).
// ---------------------------------------------------------------------------
template <int C>
__global__ __launch_bounds__(256)
void gin_scatter(const float* __restrict__ x,
                 const int*   __restrict__ srcv,
                 const int*   __restrict__ dstv,
                 float*       __restrict__ agg) {
  constexpr int QC = C / 4;                 // float4 groups per edge
  constexpr int SH = (QC == 32) ? 5 : 6;    // log2(QC) for C=128 / C=256
  static_assert((1 << SH) == QC, "C must be 128 or 256");
  const long long total  = (long long)N_EDGES * QC;
  const long long stride = (long long)gridDim.x * blockDim.x;
  long long i = (long long)blockIdx.x * blockDim.x + threadIdx.x;
  for (; i < total; i += stride) {
    const int e = (int)(i >> SH);
    const int c = ((int)i & (QC - 1)) * 4;
    const int s = srcv[e];
    const int d = dstv[e];
    const float4 v = *(const float4*)(x + (long long)s * C + c);
    float* p = agg + (long long)d * C + c;
    unsafeAtomicAdd(p + 0, v.x);
    unsafeAtomicAdd(p + 1, v.y);
    unsafeAtomicAdd(p + 2, v.z);
    unsafeAtomicAdd(p + 3, v.w);
  }
}

// ---------------------------------------------------------------------------
// Fused (1+eps)*x + agg  ->  GEMM (M=16 tile per block, N=256, K templated)
// using V_WMMA_F32_16X16X4_F32.
//
// Block = 256 threads = 8 waves. Block owns rows [m0, m0+16); wave w owns
// columns [32w, 32w+32) as two 16x16 f32 accumulators.
//
// Fragment layouts per CDNA5 ISA (CDNA5 ISA):
//   A 16x4 : lanes 0-15 row M=lane,   v0=K0 v1=K1 ; lanes 16-31 v0=K2 v1=K3
//   B 4x16 : lanes 0-15 col N=lane,   v0=K0 v1=K1 ; lanes 16-31 v0=K2 v1=K3
//   C/D    : VGPR r -> M=r (lanes 0-15) / M=r+8 (lanes 16-31), N=lane&15
// ---------------------------------------------------------------------------
template <int K, bool RELU>
__global__ __launch_bounds__(256)
void gin_linear(const float* __restrict__ xin,    // N_NODES x K
                const float* __restrict__ agg,    // N_NODES x K
                const float* __restrict__ W,      // K x 256 (row-major)
                const float* __restrict__ bias,   // 256
                const float* __restrict__ eps_p,  // scalar (1 element)
                float*       __restrict__ out) {  // N_NODES x 256
  constexpr int N   = 256;
  constexpr int LDT = K + 4;                // padded LDS stride (16B aligned)
  __shared__ float t[16 * LDT];

  const int   m0   = blockIdx.x * 16;
  const float eps1 = 1.0f + *eps_p;

  // Stage (1+eps)*x + agg for the 16-row tile into LDS (float4 granularity).
  constexpr int QK = K / 4;
  for (int idx = threadIdx.x; idx < 16 * QK; idx += 256) {
    const int r = idx / QK;
    const int k = (idx % QK) * 4;
    const long long g = (long long)(m0 + r) * K + k;
    const float4 xv = *(const float4*)(xin + g);
    const float4 av = *(const float4*)(agg + g);
    float4 tv;
    tv.x = fmaf(eps1, xv.x, av.x);
    tv.y = fmaf(eps1, xv.y, av.y);
    tv.z = fmaf(eps1, xv.z, av.z);
    tv.w = fmaf(eps1, xv.w, av.w);
    *(float4*)(t + r * LDT + k) = tv;
  }
  __syncthreads();

  const int wave = threadIdx.x >> 5;
  const int lane = threadIdx.x & 31;
  const int half = lane >> 4;               // 0: K pair {0,1}, 1: K pair {2,3}
  const int l16  = lane & 15;
  const int n0   = wave * 32;
  const int col0 = n0 + l16;
  const int col1 = n0 + 16 + l16;

  v8f acc0 = {};
  v8f acc1 = {};

  for (int k0 = 0; k0 < K; k0 += 4) {
    const int kk = k0 + half * 2;
    v2f a;
    a.x = t[l16 * LDT + kk];
    a.y = t[l16 * LDT + kk + 1];
    v2f b0, b1;
    b0.x = W[kk * N + col0];
    b0.y = W[(kk + 1) * N + col0];
    b1.x = W[kk * N + col1];
    b1.y = W[(kk + 1) * N + col1];
    acc0 = __builtin_amdgcn_wmma_f32_16x16x4_f32(
        /*neg_a=*/false, a, /*neg_b=*/false, b0,
        /*c_mod=*/(short)0, acc0, /*reuse_a=*/false, /*reuse_b=*/false);
    acc1 = __builtin_amdgcn_wmma_f32_16x16x4_f32(
        /*neg_a=*/false, a, /*neg_b=*/false, b1,
        /*c_mod=*/(short)0, acc1, /*reuse_a=*/false, /*reuse_b=*/false);
  }

  const float bias0 = bias[col0];
  const float bias1 = bias[col1];
#pragma unroll
  for (int r = 0; r < 8; r++) {
    const int m = m0 + r + half * 8;
    float v0 = acc0[r] + bias0;
    float v1 = acc1[r] + bias1;
    if (RELU) {
      v0 = fmaxf(v0, 0.0f);
      v1 = fmaxf(v1, 0.0f);
    }
    out[(long long)m * N + col0] = v0;
    out[(long long)m * N + col1] = v1;
  }
}

// ---------------------------------------------------------------------------
// Launch
// ---------------------------------------------------------------------------
extern "C" void kernel_launch(void* const* d_in, const int* in_sizes, int n_in,
                              void* d_out, int out_size, void* d_ws, size_t ws_size,
                              hipStream_t stream) {
  (void)in_sizes; (void)n_in; (void)out_size; (void)ws_size;

  const float* x    = (const float*)d_in[0];
  const int*   ei   = (const int*)d_in[1];     // [2, N_EDGES]
  const float* W1   = (const float*)d_in[2];
  const float* b1   = (const float*)d_in[3];
  const float* W2   = (const float*)d_in[4];
  const float* b2   = (const float*)d_in[5];
  const float* W3   = (const float*)d_in[6];
  const float* b3   = (const float*)d_in[7];
  const float* eps  = (const float*)d_in[8];   // 3 scalars
  float*       out  = (float*)d_out;

  const int* src = ei;
  const int* dst = ei + N_EDGES;

  const long long feat = (long long)N_NODES * HID_CH;  // 12.8M floats
  float* agg = (float*)d_ws;       // N_NODES x 256 (only first N_NODES*128 used in L1)
  float* h1  = agg + feat;         // N_NODES x 256
  float* h2  = h1 + feat;          // N_NODES x 256

  const int mblocks = N_NODES / 16;            // 3125
  const int zgrid   = 4096;

  // ---- Layer 1: C_in = 128 ----
  {
    const long long n = (long long)N_NODES * IN_CH;
    fill_zero<<<zgrid, 256, 0, stream>>>(agg, n);
    const long long work = (long long)N_EDGES * (IN_CH / 4);
    const int blocks = (int)((work + 255) / 256);
    gin_scatter<IN_CH><<<blocks, 256, 0, stream>>>(x, src, dst, agg);
    gin_linear<IN_CH, true><<<mblocks, 256, 0, stream>>>(x, agg, W1, b1, eps + 0, h1);
  }

  // ---- Layer 2: C_in = 256 ----
  {
    fill_zero<<<zgrid, 256, 0, stream>>>(agg, feat);
    const long long work = (long long)N_EDGES * (HID_CH / 4);
    const int blocks = (int)((work + 255) / 256);
    gin_scatter<HID_CH><<<blocks, 256, 0, stream>>>(h1, src, dst, agg);
    gin_linear<HID_CH, true><<<mblocks, 256, 0, stream>>>(h1, agg, W2, b2, eps + 1, h2);
  }

  // ---- Layer 3: C_in = 256, no ReLU, write d_out ----
  {
    fill_zero<<<zgrid, 256, 0, stream>>>(agg, feat);
    const long long work = (long long)N_EDGES * (HID_CH / 4);
    const int blocks = (int)((work + 255) / 256);
    gin_scatter<HID_CH><<<blocks, 256, 0, stream>>>(h2, src, dst, agg);
    gin_linear<HID_CH, false><<<mblocks, 256, 0, stream>>>(h2, agg, W3, b3, eps + 2, out);
  }
}